// AttentionHead_60413009986164
// MI455X (gfx1250) — compile-verified
//
#include <hip/hip_runtime.h>
#include <math.h>

// ---------------------------------------------------------------------------
// Causal single-head attention, all-f32 compute via V_WMMA_F32_16X16X4_F32.
//   q=x@Wq, k=x@Wk, v=x@Wv ; S = q k^T/8 + causal(-1e9) ; P=softmax(S) ; out=P v
//
// Pipeline:
//   1) qkv_proj_kernel : one wave per (16-row x-tile, matrix)   -> 3072 waves
//   2) attn_part_kernel: one wave per (batch, q-tile, split)    -> 1024*nsplit
//      split-K flash attention, unnormalized partials (O', m, l) to ws
//   3) combine_kernel  : log-sum-exp merge of the partials      -> 1024 waves
// ---------------------------------------------------------------------------

#define BS     8
#define SEQ    2048
#define D_ATTN 768
#define D_QKV  64
#define NROWS  (BS * SEQ)          // 16384
#define NTASK  (BS * (SEQ / 16))   // 1024 (batch, q-tile) tasks
#define NEGBIG (-1.0e9f)

typedef __attribute__((ext_vector_type(2))) float v2f;
typedef __attribute__((ext_vector_type(8))) float v8f;

// D(16x16,f32) = A(16x4,f32) x B(4x16,f32) + C   -- exact f32 arithmetic
__device__ __forceinline__ v8f wmma4(v2f a, v2f b, v8f c) {
  return __builtin_amdgcn_wmma_f32_16x16x4_f32(
      /*neg_a=*/false, a, /*neg_b=*/false, b,
      /*c_mod=*/(short)0, c, /*reuse_a=*/false, /*reuse_b=*/false);
}

// ---------------------------------------------------------------------------
// Kernel 1: QKV projection.  One wave per (row-tile, matrix): 3072 waves.
// The 3 waves sharing an x-tile are adjacent in the grid -> x tile is pulled
// from HBM once and re-hit in L2 (x = 50 MB << 192 MB L2).
// A-frag (16x4 f32): lane m=l&15 is the row; VGPR j holds K = 2*(l>>4)+j.
// B-frag (4x16 f32): lane n=l&15 is the col; VGPR j holds K = 2*(l>>4)+j.
// C/D  (16x16 f32): lane = col, VGPR r -> row r + 8*(l>>4).
// ---------------------------------------------------------------------------
__global__ __launch_bounds__(256) void qkv_proj_kernel(
    const float* __restrict__ x,
    const float* __restrict__ Wq, const float* __restrict__ Wk,
    const float* __restrict__ Wv,
    float* __restrict__ qkv)   // [3][NROWS][64]
{
  const int lane  = threadIdx.x & 31;
  const int wave  = threadIdx.x >> 5;
  const int flat  = blockIdx.x * 8 + wave;     // [0, 3072)
  const int which = flat % 3;                  // 0=Q 1=K 2=V
  const int rt    = flat / 3;                  // row tile [0,1024)
  const int m     = lane & 15;
  const int half  = lane >> 4;

  const float* __restrict__ W =
      (which == 0) ? Wq : ((which == 1) ? Wk : Wv);
  const float* __restrict__ xrow = x + (size_t)(rt * 16 + m) * D_ATTN;

  v8f acc[4] = {};

  for (int k0 = 0; k0 < D_ATTN; k0 += 4) {
    const int ka = k0 + 2 * half;
    v2f a; a.x = xrow[ka]; a.y = xrow[ka + 1];
#pragma unroll
    for (int nt = 0; nt < 4; ++nt) {
      const int n = nt * 16 + m;
      v2f b;
      b.x = W[(size_t)ka * D_QKV + n];
      b.y = W[(size_t)(ka + 1) * D_QKV + n];
      acc[nt] = wmma4(a, b, acc[nt]);
    }
  }

  float* __restrict__ o = qkv + (size_t)which * NROWS * D_QKV;
#pragma unroll
  for (int nt = 0; nt < 4; ++nt)
#pragma unroll
    for (int r = 0; r < 8; ++r)
      o[(size_t)(rt * 16 + r + 8 * half) * D_QKV + nt * 16 + m] = acc[nt][r];
}

// ---------------------------------------------------------------------------
// Kernel 2: split-K flash attention partials.
// One wave per (batch, q-tile, split); split s handles kt = s, s+nsplit, ...
// S^T = K_tile(16k x 64d) x Q^T(64d x 16q): softmax rows (q) land in lanes,
// so row reductions are 8 per-lane ops + one __shfl_xor(.,16).
// Writes UNNORMALIZED O' plus (m, l) per q row.
// ---------------------------------------------------------------------------
__global__ __launch_bounds__(256) void attn_part_kernel(
    const float* __restrict__ qkv,
    float* __restrict__ Opart,   // [NTASK*nsplit][16][64]
    float* __restrict__ mpart,   // [NTASK*nsplit][16]
    float* __restrict__ lpart,   // [NTASK*nsplit][16]
    int nsplit)
{
  const int lane  = threadIdx.x & 31;
  const int wave  = threadIdx.x >> 5;
  const int part  = blockIdx.x * 8 + wave;     // [0, NTASK*nsplit)
  const int task  = part / nsplit;             // (b, qt)
  const int split = part % nsplit;
  const int b     = task >> 7;
  const int qt    = task & 127;
  const int m     = lane & 15;
  const int half  = lane >> 4;

  const float* __restrict__ Q = qkv + (size_t)b * SEQ * D_QKV;
  const float* __restrict__ K = qkv + ((size_t)NROWS + (size_t)b * SEQ) * D_QKV;
  const float* __restrict__ V = qkv + (2 * (size_t)NROWS + (size_t)b * SEQ) * D_QKV;

  // Preload Q tile as 16 B-fragments (B element (K=d, N=q) = Q[q][d]).
  v2f qf[16];
  {
    const float* __restrict__ qrow = Q + (size_t)(qt * 16 + m) * D_QKV;
#pragma unroll
    for (int s = 0; s < 16; ++s) {
      const int d = s * 4 + 2 * half;
      qf[s].x = qrow[d]; qf[s].y = qrow[d + 1];
    }
  }

  v8f   O[4] = {};
  float mrow = -INFINITY;   // running max for q = m (replicated in both halves)
  float lrow = 0.0f;        // running sum for q = m
  const float scale = 0.125f;   // 1/sqrt(64)
  const int   qidx  = qt * 16 + m;

  for (int kt = split; kt <= qt; kt += nsplit) {
    // ---- S^T tile: M = local k row, N = local q col --------------------
    v8f S = {};
    const float* __restrict__ krow = K + (size_t)(kt * 16 + m) * D_QKV;
#pragma unroll
    for (int s = 0; s < 16; ++s) {
      const int d = s * 4 + 2 * half;
      v2f a; a.x = krow[d]; a.y = krow[d + 1];
      S = wmma4(a, qf[s], S);
    }

    // ---- scale + causal mask + tile row max ----------------------------
    float tmax = -INFINITY;
#pragma unroll
    for (int r = 0; r < 8; ++r) {
      float sv = S[r] * scale;
      const int kidx = kt * 16 + r + 8 * half;
      if (kidx > qidx) sv += NEGBIG;      // reproduces mask = -1e9 above diag
      S[r] = sv;
      tmax = fmaxf(tmax, sv);
    }
    tmax = fmaxf(tmax, __shfl_xor(tmax, 16, 32));  // combine k halves

    const float mnew    = fmaxf(mrow, tmax);
    const float rescale = __expf(mrow - mnew);     // exp(-inf)=0 on first tile
    mrow = mnew;

    float tsum = 0.0f;
#pragma unroll
    for (int r = 0; r < 8; ++r) {
      const float p = __expf(S[r] - mnew);
      S[r] = p;
      tsum += p;
    }
    tsum += __shfl_xor(tsum, 16, 32);
    lrow = lrow * rescale + tsum;

    // ---- rescale O: reg r belongs to q = r + 8*half --------------------
#pragma unroll
    for (int r = 0; r < 8; ++r) {
      const float sc = __shfl(rescale, r + 8 * half, 32);
#pragma unroll
      for (int nt = 0; nt < 4; ++nt) O[nt][r] *= sc;
    }

    // ---- O += P x V : convert P^T (C-layout) to A-frags via one
    //      cross-half shuffle per register, then 4x4 WMMAs ----------------
#pragma unroll
    for (int s = 0; s < 4; ++s) {        // K-step over local k (4 per step)
      v2f a;
#pragma unroll
      for (int j = 0; j < 2; ++j) {
        float av;
        if (s < 2) {
          // needed k < 8 for both halves: source is half0's registers
          const float v_own = S[s * 4 + j];                         // half0
          const float v_oth = __shfl_xor(S[s * 4 + j + 2], 16, 32); // half1
          av = half ? v_oth : v_own;
        } else {
          // needed k >= 8 for both halves: source is half1's registers
          const float v_own = S[s * 4 + j - 6];                     // half1
          const float v_oth = __shfl_xor(S[s * 4 + j - 8], 16, 32); // half0
          av = half ? v_own : v_oth;
        }
        if (j == 0) a.x = av; else a.y = av;
      }
      const int kk = kt * 16 + s * 4 + 2 * half;   // V row for this lane/frag
#pragma unroll
      for (int nt = 0; nt < 4; ++nt) {
        v2f bv;
        bv.x = V[(size_t)kk * D_QKV + nt * 16 + m];
        bv.y = V[(size_t)(kk + 1) * D_QKV + nt * 16 + m];
        O[nt] = wmma4(a, bv, O[nt]);
      }
    }
  }

  // ---- store unnormalized partial (O', m, l) ---------------------------
#pragma unroll
  for (int nt = 0; nt < 4; ++nt)
#pragma unroll
    for (int r = 0; r < 8; ++r)
      Opart[((size_t)part * 16 + r + 8 * half) * D_QKV + nt * 16 + m] = O[nt][r];
  if (half == 0) {               // q = m stats live identically in both halves
    mpart[(size_t)part * 16 + m] = mrow;
    lpart[(size_t)part * 16 + m] = lrow;
  }
}

// ---------------------------------------------------------------------------
// Kernel 3: merge split-K partials.  One wave per (b, q-tile).
//   M = max_s m_s ;  out = (sum_s e^{m_s-M} O'_s) / (sum_s e^{m_s-M} l_s)
// ---------------------------------------------------------------------------
__global__ __launch_bounds__(256) void combine_kernel(
    const float* __restrict__ Opart, const float* __restrict__ mpart,
    const float* __restrict__ lpart, float* __restrict__ out, int nsplit)
{
  const int lane = threadIdx.x & 31;
  const int wave = threadIdx.x >> 5;
  const int task = blockIdx.x * 8 + wave;      // [0, NTASK)
  const int b    = task >> 7;
  const int qt   = task & 127;

  for (int q = 0; q < 16; ++q) {
    float M = -INFINITY;
    for (int s = 0; s < nsplit; ++s)
      M = fmaxf(M, mpart[(size_t)(task * nsplit + s) * 16 + q]);
    float denom = 0.0f, acc0 = 0.0f, acc1 = 0.0f;
    for (int s = 0; s < nsplit; ++s) {
      const size_t ps = (size_t)(task * nsplit + s);
      const float  w  = __expf(mpart[ps * 16 + q] - M);  // 0 for empty splits
      denom += w * lpart[ps * 16 + q];
      const float* __restrict__ Ob = Opart + (ps * 16 + q) * D_QKV;
      acc0 += w * Ob[lane];
      acc1 += w * Ob[lane + 32];
    }
    const float  inv = 1.0f / denom;
    const size_t row = (size_t)b * SEQ + qt * 16 + q;
    out[row * D_QKV + lane]      = acc0 * inv;
    out[row * D_QKV + lane + 32] = acc1 * inv;
  }
}

// ---------------------------------------------------------------------------
extern "C" void kernel_launch(void* const* d_in, const int* in_sizes, int n_in,
                              void* d_out, int out_size, void* d_ws, size_t ws_size,
                              hipStream_t stream) {
  const float* x   = (const float*)d_in[0];
  // d_in[1] = mask: causal structure reproduced analytically, not read.
  const float* Wq  = (const float*)d_in[2];
  const float* Wk  = (const float*)d_in[3];
  const float* Wv  = (const float*)d_in[4];
  float*       out = (float*)d_out;
  float*       ws  = (float*)d_ws;

  const size_t qkv_f = 3ull * NROWS * D_QKV;        // 12.6 MB
  // per (task,split): 16x64 O' + 16 m + 16 l = 1056 floats
  int nsplit = 4;
  while (nsplit > 1 &&
         ws_size < (qkv_f + (size_t)nsplit * NTASK * 1056) * sizeof(float))
    nsplit >>= 1;

  float* qkv   = ws;
  float* Opart = qkv + qkv_f;
  float* mpart = Opart + (size_t)nsplit * NTASK * 16 * D_QKV;
  float* lpart = mpart + (size_t)nsplit * NTASK * 16;

  // 1024 row tiles x 3 matrices = 3072 waves (8 waves / block)
  qkv_proj_kernel<<<384, 256, 0, stream>>>(x, Wq, Wk, Wv, qkv);
  // 1024 (b, q-tile) tasks x nsplit = up to 4096 balanced waves
  attn_part_kernel<<<128 * nsplit, 256, 0, stream>>>(qkv, Opart, mpart, lpart, nsplit);
  // 1024 merge waves
  combine_kernel<<<128, 256, 0, stream>>>(Opart, mpart, lpart, out, nsplit);
}